// Decoder_48112223650119
// MI455X (gfx1250) — compile-verified
//
#include <hip/hip_runtime.h>

#define BB   16
#define TT   256
#define IDIMK 512
#define ODIMK 80
#define LLK  100
#define DU   1024
#define PUK  256
#define ADK  128
#define ACHK 32
#define AKK  31
#define PCHK 512
#define PKK  5
#define IN0  (IDIMK + PUK)   // 768
#define NWG  64
#define NTHR 256
#define LTILES 7             // ceil(LLK/16)

typedef __attribute__((ext_vector_type(2))) float v2f;
typedef __attribute__((ext_vector_type(8))) float v8f;

struct Params {
  const float *hs; const int *hlens; const float *ys;
  const float *W_pre0, *W_pre1;
  const float *W_ih0, *W_hh0, *b_ih0, *b_hh0;
  const float *W_ih1, *W_hh1, *b_ih1, *b_hh1;
  const float *W_enc, *b_enc, *W_dec, *W_att, *loc_conv, *gvec, *g_b;
  const float *W_feat, *W_prob, *b_prob;
  const float *pw[5], *pg[5], *pb[5];
  float *after_, *before_, *logits_, *attws_;
  float *enc_h, *encWt, *attWt, *conv, *dec, *e, *attc, *p0, *p1;
  float *z0[2], *c0[2], *z1[2], *c1[2], *cum[2];
  float *pnA, *pnB;
  int *sync;
};

__device__ __forceinline__ float sigmoidf_(float x) { return 1.0f / (1.0f + __expf(-x)); }

// ---------------- device-wide barrier (persistent kernel) ----------------
__device__ __forceinline__ void gsync(int *syncp) {
  __syncthreads();
  if (threadIdx.x == 0) {
    __threadfence();
    volatile int *gen = syncp + 1;
    int g = *gen;
    if (atomicAdd(syncp, 1) == (int)gridDim.x - 1) {
      syncp[0] = 0;
      __threadfence();
      atomicAdd((int *)(syncp + 1), 1);
    } else {
      while (*gen == g) { __builtin_amdgcn_s_sleep(2); }
    }
    __threadfence();
  }
  __syncthreads();
}

// ---------------- f32 WMMA 16x16x4 K-loop ----------
// D(16x16) += A(16xK) * W(16 rows x K)^T.  Lane r=lane&15 is both the A-row
// (M) and the weight-row (N); kk=(lane>>4)*2 selects the K sub-pair.
__device__ __forceinline__ v8f wmma_span(v8f acc, const float *A, int lda,
                                         const float *W, int ldw, int K) {
  const int lane = threadIdx.x & 31;
  const int r  = lane & 15;
  const int kk = (lane >> 4) << 1;
  const float *ap = A + (long)r * lda + kk;
  const float *bp = W + (long)r * ldw + kk;
#pragma unroll 8
  for (int k = 0; k < K; k += 4) {
    v2f a; a.x = ap[k]; a.y = ap[k + 1];
    v2f b; b.x = bp[k]; b.y = bp[k + 1];
    acc = __builtin_amdgcn_wmma_f32_16x16x4_f32(false, a, false, b,
                                                (short)0, acc, false, false);
  }
  return acc;
}

// ---------------- one LSTM cell step for a 16-column group -------------
__device__ void lstm_block(int j0,
                           const float *A0a, int ld0a, int K0a,
                           const float *A0b, int ld0b, int K0b,
                           const float *Wih, int ldih,
                           const float *A1, int ld1, int K1,
                           const float *Whh, int ldhh,
                           const float *bih, const float *bhh,
                           const float *zprev, const float *cprev,
                           float *znew, float *cnew, float *smem) {
  const int wave = threadIdx.x >> 5;
  const int lane = threadIdx.x & 31;
  const int gate = wave & 3;
  const int half = wave >> 2;
  const int rowbase = gate * DU + j0;
  v8f acc = {};
  if (half == 0) {
    acc = wmma_span(acc, A0a, ld0a, Wih + (long)rowbase * ldih, ldih, K0a);
    if (A0b)
      acc = wmma_span(acc, A0b, ld0b, Wih + (long)rowbase * ldih + K0a, ldih, K0b);
  } else {
    acc = wmma_span(acc, A1, ld1, Whh + (long)rowbase * ldhh, ldhh, K1);
  }
  const int colg = lane & 15;
  const int rbase = (lane >> 4) * 8;
  float *tb = smem + wave * 256;
#pragma unroll
  for (int v = 0; v < 8; ++v) tb[(rbase + v) * 16 + colg] = acc[v];
  __syncthreads();
  {
    const int t = threadIdx.x;            // 16 batch x 16 cols
    const int m = t >> 4;
    const int j = j0 + (t & 15);
    float gi = smem[0 * 256 + t] + smem[4 * 256 + t] + bih[0 * DU + j] + bhh[0 * DU + j];
    float gf = smem[1 * 256 + t] + smem[5 * 256 + t] + bih[1 * DU + j] + bhh[1 * DU + j];
    float gg = smem[2 * 256 + t] + smem[6 * 256 + t] + bih[2 * DU + j] + bhh[2 * DU + j];
    float go = smem[3 * 256 + t] + smem[7 * 256 + t] + bih[3 * DU + j] + bhh[3 * DU + j];
    float cold = cprev[m * DU + j];
    float c2 = sigmoidf_(gf) * cold + sigmoidf_(gi) * tanhf(gg);
    float h  = sigmoidf_(go) * tanhf(c2);
    znew[m * DU + j] = 0.1f * zprev[m * DU + j] + 0.9f * h;   // zoneout
    cnew[m * DU + j] = 0.1f * cold + 0.9f * c2;
  }
  __syncthreads();
}

// ---------------- output projection + stop logit for one step ----------
__device__ void out_proj(const Params &P, int s, const float *z1v,
                         int tid, int nthr) {
  for (int idx = tid; idx < BB * ODIMK; idx += nthr) {
    int b = idx / ODIMK, o = idx % ODIMK;
    const float *zr = z1v + b * DU;
    const float *cr = P.attc + b * IDIMK;
    float sum = 0.f;
    for (int k = 0; k < DU; ++k) sum += zr[k] * P.W_feat[k * ODIMK + o];
    for (int k = 0; k < IDIMK; ++k) sum += cr[k] * P.W_feat[(DU + k) * ODIMK + o];
    P.before_[((long)b * LLK + s) * ODIMK + o] = sum;
  }
  if (tid < BB) {
    int b = tid;
    const float *zr = z1v + b * DU;
    const float *cr = P.attc + b * IDIMK;
    float sum = P.b_prob[0];
    for (int k = 0; k < DU; ++k) sum += zr[k] * P.W_prob[k];
    for (int k = 0; k < IDIMK; ++k) sum += cr[k] * P.W_prob[DU + k];
    P.logits_[b * LLK + s] = sum;
  }
}

// ---------------- postnet conv layer as WMMA GEMM ----------------------
__device__ __forceinline__ void post_gemm(const float *in, int inIsBLC,
                                          int Cin, int Cout,
                                          const float *w, const float *g,
                                          const float *bb, int act,
                                          float *out, float *smem) {
  const int K = Cin * PKK;
  const int CB = (Cout + 127) / 128;        // col-blocks of 8 waves x 16
  const int njobs = BB * LTILES * CB;
  const int wave = threadIdx.x >> 5;
  const int lane = threadIdx.x & 31;
  const float bn = 0.9999950000374997f;     // 1/sqrt(1+1e-5)
  for (int job = blockIdx.x; job < njobs; job += NWG) {
    int mt = job % (BB * LTILES);
    int cb = job / (BB * LTILES);
    int b = mt / LTILES, l0 = (mt % LTILES) * 16;
    int co0 = cb * 128 + wave * 16;
    v8f acc = {};
    for (int k0 = 0; k0 < K; k0 += 128) {
      int kc = (K - k0 < 128) ? (K - k0) : 128;
      __syncthreads();
      for (int i = threadIdx.x; i < 16 * 128; i += NTHR) {
        int row = i >> 7, kkc = i & 127;
        float v = 0.f;
        int gk = k0 + kkc;
        int l = l0 + row;
        if (kkc < kc && l < LLK) {
          int ci = gk / PKK, p = gk % PKK;
          int ll = l + p - 2;
          if (ll >= 0 && ll < LLK)
            v = inIsBLC ? in[(b * LLK + ll) * ODIMK + ci]
                        : in[((long)b * Cin + ci) * LLK + ll];
        }
        smem[row * 132 + kkc] = v;
      }
      __syncthreads();
      if (co0 < Cout) {
        const int r = lane & 15;
        const int kk = (lane >> 4) << 1;
        const float *ap = smem + r * 132 + kk;
        const float *bp = w + (long)(co0 + r) * K + k0 + kk;
#pragma unroll 4
        for (int k = 0; k < kc; k += 4) {
          v2f a; a.x = ap[k]; a.y = ap[k + 1];
          v2f bv; bv.x = bp[k]; bv.y = bp[k + 1];
          acc = __builtin_amdgcn_wmma_f32_16x16x4_f32(false, a, false, bv,
                                                      (short)0, acc, false, false);
        }
      }
    }
    if (co0 < Cout) {
      int col = lane & 15;
      int co = co0 + col;
      float gv = g[co], bbv = bb[co];
      int rbase = (lane >> 4) * 8;
#pragma unroll
      for (int v = 0; v < 8; ++v) {
        int l = l0 + rbase + v;
        if (l < LLK) {
          float y = gv * (acc[v] * bn) + bbv;
          if (act) y = tanhf(y);
          out[((long)b * Cout + co) * LLK + l] = y;
        }
      }
    }
  }
}

__global__ __launch_bounds__(NTHR, 1) void decoder_persist_kernel(Params P) {
  __shared__ float smem[2304];
  const int tid = blockIdx.x * NTHR + threadIdx.x;
  const int nthr = gridDim.x * NTHR;
  const int lane = threadIdx.x & 31;
  const int wave = threadIdx.x >> 5;

  // ---- init: transpose W_enc and W_att; zero recurrent state ----
  for (int idx = tid; idx < ADK * IDIMK; idx += nthr) {
    int a = idx / IDIMK, i = idx % IDIMK;
    P.encWt[idx] = P.W_enc[i * ADK + a];
  }
  for (int idx = tid; idx < ADK * ACHK; idx += nthr) {
    int a = idx / ACHK, c = idx % ACHK;
    P.attWt[idx] = P.W_att[c * ADK + a];
  }
  for (int idx = tid; idx < BB * DU; idx += nthr) {
    P.z0[0][idx] = 0.f; P.z0[1][idx] = 0.f;
    P.c0[0][idx] = 0.f; P.c0[1][idx] = 0.f;
    P.z1[0][idx] = 0.f; P.z1[1][idx] = 0.f;
    P.c1[0][idx] = 0.f; P.c1[1][idx] = 0.f;
  }
  for (int idx = tid; idx < BB * TT; idx += nthr) {
    P.cum[0][idx] = 0.f; P.cum[1][idx] = 0.f;
  }
  gsync(P.sync);

  // ---- enc_h = hs @ W_enc + b_enc, as WMMA GEMM (4096x512 -> 128) ----
  {
    const int gwave = blockIdx.x * 8 + wave;
    for (int job = gwave; job < 256 * 8; job += NWG * 8) {
      int mtile = job & 255;
      int a0 = (job >> 8) * 16;
      v8f acc = {};
      acc = wmma_span(acc, P.hs + (long)mtile * 16 * IDIMK, IDIMK,
                      P.encWt + (long)a0 * IDIMK, IDIMK, IDIMK);
      int col = lane & 15, rbase = (lane >> 4) * 8;
      float be = P.b_enc[a0 + col];
#pragma unroll
      for (int v = 0; v < 8; ++v)
        P.enc_h[(long)(mtile * 16 + rbase + v) * ADK + a0 + col] = acc[v] + be;
    }
  }
  gsync(P.sync);

  // ======================= sequential decoder scan =======================
  // 5 device-wide barriers per step.
  for (int step = 0; step < LLK; ++step) {
    const int pp = step & 1, qq = pp ^ 1;

    // -- phase 1: (deferred output projection of step-1) + location conv
    //             ((b,t,c) layout, coalesced) + dec = z0@W_dec + prenet L0 --
    if (step > 0) out_proj(P, step - 1, P.z1[pp], tid, nthr);
    for (int idx = tid; idx < BB * TT * ACHK; idx += nthr) {
      int b = idx / (TT * ACHK);
      int t = (idx / ACHK) % TT;
      int c = idx % ACHK;
      int hl = P.hlens[b];
      const float *awrow = P.cum[pp] + b * TT;
      float s = 0.f;
      for (int k = 0; k < AKK; ++k) {
        int t2 = t + k - (AKK - 1) / 2;
        if (t2 >= 0 && t2 < TT) {
          float a = (step == 0) ? ((t2 < hl) ? 1.0f / (float)hl : 0.f) : awrow[t2];
          s += a * P.loc_conv[c * AKK + k];
        }
      }
      P.conv[idx] = s;    // (b, t, c)
    }
    for (int idx = tid; idx < BB * ADK; idx += nthr) {
      int b = idx / ADK, a = idx % ADK;
      const float *zr = P.z0[pp] + b * DU;
      float s = 0.f;
      for (int k = 0; k < DU; ++k) s += zr[k] * P.W_dec[k * ADK + a];
      P.dec[idx] = s;
    }
    for (int idx = tid; idx < BB * PUK; idx += nthr) {
      int b = idx / PUK, u = idx % PUK;
      float s = 0.f;
      if (step > 0) {
        const float *pr = P.ys + ((long)b * LLK + (step - 1)) * ODIMK;
        for (int k = 0; k < ODIMK; ++k) s += pr[k] * P.W_pre0[k * PUK + u];
      }
      P.p0[idx] = fmaxf(s, 0.f);
    }
    gsync(P.sync);

    // -- phase 2: prenet L1 (VALU) + attention energies via WMMA:
    //    per 16-row (b,t) block: feat = conv@W_att^T (8 waves x 8 wmma,
    //    K=32), then tanh(feat+enc_h+dec).gvec reduced in LDS -> e --
    for (int idx = tid; idx < BB * PUK; idx += nthr) {
      int b = idx / PUK, u = idx % PUK;
      const float *pr = P.p0 + b * PUK;
      float s = 0.f;
      for (int k = 0; k < PUK; ++k) s += pr[k] * P.W_pre1[k * PUK + u];
      P.p1[idx] = fmaxf(s, 0.f);
    }
    for (int job = blockIdx.x; job < (BB * TT) / 16; job += NWG) {
      const int bt0 = job * 16;
      const int a0 = wave * 16;
      v8f acc = {};
      acc = wmma_span(acc, P.conv + (long)bt0 * ACHK, ACHK,
                      P.attWt + (long)a0 * ACHK, ACHK, ACHK);
      const int col = lane & 15;
      const int a = a0 + col;
      const int rbase = (lane >> 4) * 8;
      const float gv = P.gvec[a];
      float *tb = smem + wave * 256;
      __syncthreads();
#pragma unroll
      for (int v = 0; v < 8; ++v) {
        int bt = bt0 + rbase + v;
        int b = bt >> 8;
        float sf = acc[v] + P.enc_h[(long)bt * ADK + a] + P.dec[b * ADK + a];
        tb[(rbase + v) * 16 + col] = tanhf(sf) * gv;
      }
      __syncthreads();
      {  // two-stage reduction over the 128 a-columns
        int t = threadIdx.x;
        int row = t >> 4, seg = t & 15;
        float part = 0.f;
#pragma unroll
        for (int w8 = 0; w8 < 8; ++w8) part += smem[w8 * 256 + row * 16 + seg];
        smem[2048 + t] = part;
      }
      __syncthreads();
      if (threadIdx.x < 16) {
        float s = P.g_b[0];
#pragma unroll
        for (int seg = 0; seg < 16; ++seg) s += smem[2048 + threadIdx.x * 16 + seg];
        P.e[bt0 + threadIdx.x] = s;
      }
      __syncthreads();
    }
    gsync(P.sync);

    // -- phase 3 (fused): masked scaled softmax + cumulate + context.
    //    WG b owns batch row b; the context sum reads aw from LDS --
    if (blockIdx.x < BB) {
      int b = blockIdx.x, t = threadIdx.x;
      int hl = P.hlens[b];
      float val = (t < hl) ? 2.0f * P.e[b * TT + t] : -3.0e38f;
      smem[t] = val; __syncthreads();
      for (int off = 128; off; off >>= 1) {
        if (t < off) smem[t] = fmaxf(smem[t], smem[t + off]);
        __syncthreads();
      }
      float mx = smem[0]; __syncthreads();
      float ex = (t < hl) ? __expf(val - mx) : 0.f;
      smem[t] = ex; __syncthreads();
      for (int off = 128; off; off >>= 1) {
        if (t < off) smem[t] += smem[t + off];
        __syncthreads();
      }
      float a = ex / smem[0];
      P.attws_[((long)b * LLK + step) * TT + t] = a;
      P.cum[qq][b * TT + t] = (step == 0) ? a : P.cum[pp][b * TT + t] + a;
      __syncthreads();
      smem[t] = a;            // share the aw row inside the WG
      __syncthreads();
      for (int i = t; i < IDIMK; i += NTHR) {   // att_c[b,i] = sum_t aw.hs
        const float *hr = P.hs + (long)b * TT * IDIMK + i;
        float s = 0.f;
        for (int tt = 0; tt < TT; ++tt) s += smem[tt] * hr[(long)tt * IDIMK];
        P.attc[b * IDIMK + i] = s;
      }
    }
    gsync(P.sync);

    // -- phase 4: LSTM0 (WMMA): x = [att_c | prenet], h = z0 --
    lstm_block(blockIdx.x * 16,
               P.attc, IDIMK, IDIMK, P.p1, PUK, PUK, P.W_ih0, IN0,
               P.z0[pp], DU, DU, P.W_hh0, DU,
               P.b_ih0, P.b_hh0, P.z0[pp], P.c0[pp], P.z0[qq], P.c0[qq], smem);
    gsync(P.sync);

    // -- phase 5: LSTM1 (WMMA): x = z0_new, h = z1 --
    lstm_block(blockIdx.x * 16,
               P.z0[qq], DU, DU, (const float *)0, 0, 0, P.W_ih1, DU,
               P.z1[pp], DU, DU, P.W_hh1, DU,
               P.b_ih1, P.b_hh1, P.z1[pp], P.c1[pp], P.z1[qq], P.c1[qq], smem);
    gsync(P.sync);
  }

  // final step's output projection (states live in buffer index LLK&1 == 0)
  out_proj(P, LLK - 1, P.z1[LLK & 1], tid, nthr);
  gsync(P.sync);

  // ======================= postnet (WMMA GEMM layers) =======================
  post_gemm(P.before_, 1, ODIMK, PCHK, P.pw[0], P.pg[0], P.pb[0], 1, P.pnA, smem);
  gsync(P.sync);
  post_gemm(P.pnA, 0, PCHK, PCHK, P.pw[1], P.pg[1], P.pb[1], 1, P.pnB, smem);
  gsync(P.sync);
  post_gemm(P.pnB, 0, PCHK, PCHK, P.pw[2], P.pg[2], P.pb[2], 1, P.pnA, smem);
  gsync(P.sync);
  post_gemm(P.pnA, 0, PCHK, PCHK, P.pw[3], P.pg[3], P.pb[3], 1, P.pnB, smem);
  gsync(P.sync);
  post_gemm(P.pnB, 0, PCHK, ODIMK, P.pw[4], P.pg[4], P.pb[4], 0, P.pnA, smem);
  gsync(P.sync);
  for (int idx = tid; idx < BB * LLK * ODIMK; idx += nthr) {
    int b = idx / (LLK * ODIMK);
    int l = (idx / ODIMK) % LLK;
    int o = idx % ODIMK;
    P.after_[idx] = P.before_[idx] + P.pnA[((long)b * ODIMK + o) * LLK + l];
  }
}

extern "C" void kernel_launch(void *const *d_in, const int *in_sizes, int n_in,
                              void *d_out, int out_size, void *d_ws, size_t ws_size,
                              hipStream_t stream) {
  Params P;
  P.hs = (const float *)d_in[0];
  P.hlens = (const int *)d_in[1];
  P.ys = (const float *)d_in[2];
  P.W_pre0 = (const float *)d_in[3];
  P.W_pre1 = (const float *)d_in[4];
  P.W_ih0 = (const float *)d_in[5];
  P.W_hh0 = (const float *)d_in[6];
  P.b_ih0 = (const float *)d_in[7];
  P.b_hh0 = (const float *)d_in[8];
  P.W_ih1 = (const float *)d_in[9];
  P.W_hh1 = (const float *)d_in[10];
  P.b_ih1 = (const float *)d_in[11];
  P.b_hh1 = (const float *)d_in[12];
  P.W_enc = (const float *)d_in[13];
  P.b_enc = (const float *)d_in[14];
  P.W_dec = (const float *)d_in[15];
  P.W_att = (const float *)d_in[16];
  P.loc_conv = (const float *)d_in[17];
  P.gvec = (const float *)d_in[18];
  P.g_b = (const float *)d_in[19];
  P.W_feat = (const float *)d_in[20];
  P.W_prob = (const float *)d_in[21];
  P.b_prob = (const float *)d_in[22];
  for (int i = 0; i < 5; ++i) {
    P.pw[i] = (const float *)d_in[23 + i];
    P.pg[i] = (const float *)d_in[28 + 2 * i];
    P.pb[i] = (const float *)d_in[29 + 2 * i];
  }
  float *o = (float *)d_out;
  P.after_ = o;
  P.before_ = o + (long)BB * LLK * ODIMK;
  P.logits_ = P.before_ + (long)BB * LLK * ODIMK;
  P.attws_ = P.logits_ + (long)BB * LLK;

  float *w = (float *)d_ws;
  P.sync = (int *)w;           w += 64;
  P.enc_h = w;                 w += (long)BB * TT * ADK;
  P.encWt = w;                 w += (long)ADK * IDIMK;
  P.attWt = w;                 w += (long)ADK * ACHK;
  P.conv = w;                  w += (long)BB * TT * ACHK;
  P.dec = w;                   w += BB * ADK;
  P.e = w;                     w += BB * TT;
  P.attc = w;                  w += BB * IDIMK;
  P.p0 = w;                    w += BB * PUK;
  P.p1 = w;                    w += BB * PUK;
  for (int i = 0; i < 2; ++i) { P.z0[i] = w; w += BB * DU; }
  for (int i = 0; i < 2; ++i) { P.c0[i] = w; w += BB * DU; }
  for (int i = 0; i < 2; ++i) { P.z1[i] = w; w += BB * DU; }
  for (int i = 0; i < 2; ++i) { P.c1[i] = w; w += BB * DU; }
  for (int i = 0; i < 2; ++i) { P.cum[i] = w; w += BB * TT; }
  P.pnA = w;                   w += (long)BB * PCHK * LLK;
  P.pnB = w;                   w += (long)BB * PCHK * LLK;

  hipMemsetAsync(d_ws, 0, 256, stream);   // zero barrier counters (capture-safe)
  hipLaunchKernelGGL(decoder_persist_kernel, dim3(NWG), dim3(NTHR), 0, stream, P);
}